// ChamferLossOverlap_66623532696128
// MI455X (gfx1250) — compile-verified
//
#include <hip/hip_runtime.h>

#define B 8
#define N 4096
#define BN (B * N)
#define TILES (N / 16)          // 256 column tiles
#define CHUNKS 4
#define TILES_PER_CHUNK (TILES / CHUNKS)

typedef __attribute__((ext_vector_type(2))) float v2f;
typedef __attribute__((ext_vector_type(8))) float v8f;

// Monotonic float <-> uint key: unsigned compare order == float compare order.
__device__ __forceinline__ unsigned f2key(float f) {
  unsigned u = __float_as_uint(f);
  return (u & 0x80000000u) ? ~u : (u | 0x80000000u);
}
__device__ __forceinline__ float key2f(unsigned k) {
  unsigned u = (k & 0x80000000u) ? (k & 0x7FFFFFFFu) : ~k;
  return __uint_as_float(u);
}

// Squared norms of both point sets + init min-keys to +inf.
__global__ __launch_bounds__(256) void prep_kernel(
    const float* __restrict__ x, const float* __restrict__ y,
    float* __restrict__ sqx, float* __restrict__ sqy,
    unsigned* __restrict__ keys /* 2*BN */) {
  int i = blockIdx.x * blockDim.x + threadIdx.x;
  if (i < BN) {
    float x0 = x[3 * i], x1 = x[3 * i + 1], x2 = x[3 * i + 2];
    sqx[i] = x0 * x0 + x1 * x1 + x2 * x2;
    float y0 = y[3 * i], y1 = y[3 * i + 1], y2 = y[3 * i + 2];
    sqy[i] = y0 * y0 + y1 * y1 + y2 * y2;
    keys[i] = 0xFFFFFFFFu;        // dir 0: min over y-index, per x row
    keys[i + BN] = 0xFFFFFFFFu;   // dir 1: min over x-index, per y row
  }
}

// One wave = one 16-row tile of A, one chunk of column tiles of Bp.
// K=4 trick: A = [-2*a0, -2*a1, -2*a2, 1], B = [b0, b1, b2, |b_j|^2], C = 0
//   => D[i,j] = |b_j|^2 - 2 a_i.b_j  in a single WMMA, no C splat needed.
// Running min kept in the 8-VGPR accumulator layout, then lane-reduce + atomic-min.
__global__ __launch_bounds__(32) void rowmin_kernel(
    const float* __restrict__ x, const float* __restrict__ y,
    const float* __restrict__ sqx, const float* __restrict__ sqy,
    unsigned* __restrict__ keys) {
  const int tile_i = blockIdx.x;
  const int chunk  = blockIdx.y;
  const int bz     = blockIdx.z;
  const int b      = bz >> 1;
  const int dir    = bz & 1;

  const float* Ap  = dir ? y : x;
  const float* Bp  = dir ? x : y;
  const float* sqA = dir ? sqy : sqx;
  const float* sqB = dir ? sqx : sqy;
  unsigned* out    = keys + (dir ? BN : 0);

  const int  lane = threadIdx.x;
  const int  m    = lane & 15;
  const bool hi   = lane >= 16;
  const int  kofs = hi ? 2 : 0;   // which element feeds VGPR0 (K=0 low / K=2 high)

  // A fragment (16x4 f32): VGPR0 = K0(low)/K2(high), VGPR1 = K1(low)/K3(high).
  // All loads unconditional (always in-bounds), selection via cndmask — no branches.
  const int base = b * N;
  const int rowg = base + tile_i * 16 + m;
  const float a_v0 = Ap[3 * rowg + kofs];
  const float a_v1 = Ap[3 * rowg + 1];
  v2f afrag;
  afrag.x = -2.0f * a_v0;
  afrag.y = hi ? 1.0f : (-2.0f * a_v1);   // K=3 slot carries the "+1 * |b|^2" row

  const v8f zeroC = {0.f, 0.f, 0.f, 0.f, 0.f, 0.f, 0.f, 0.f};

  const float INF = __builtin_inff();
  v8f acc;
#pragma unroll
  for (int k = 0; k < 8; ++k) acc[k] = INF;

  // Fixed-stride pointers: 16 columns * 3 floats = 48 floats per tile step.
  const int colg0 = base + chunk * TILES_PER_CHUNK * 16 + m;
  const float* bp0 = Bp + 3 * colg0 + kofs;  // feeds VGPR0 (K0 / K2)
  const float* bp1 = Bp + 3 * colg0 + 1;     // feeds VGPR1 low (K1)
  const float* sp  = sqB + colg0;            // feeds VGPR1 high (K3 = |b_j|^2)

  for (int jt = 0; jt < TILES_PER_CHUNK; ++jt) {
    const float b_v0 = bp0[jt * 48];
    const float b_v1 = bp1[jt * 48];
    const float sy   = sp[jt * 16];
    v2f bfrag;
    bfrag.x = b_v0;
    bfrag.y = hi ? sy : b_v1;

    v8f d = __builtin_amdgcn_wmma_f32_16x16x4_f32(
        /*neg_a=*/false, afrag, /*neg_b=*/false, bfrag,
        /*c_mod=*/(short)0, zeroC, /*reuse_a=*/false, /*reuse_b=*/false);

#pragma unroll
    for (int k = 0; k < 8; ++k) acc[k] = fminf(acc[k], d[k]);
  }

  // Min across the 16 lanes of each half (xor masks < 16 stay within halves).
#pragma unroll
  for (int k = 0; k < 8; ++k) {
    float v = acc[k];
    v = fminf(v, __shfl_xor(v, 1, 32));
    v = fminf(v, __shfl_xor(v, 2, 32));
    v = fminf(v, __shfl_xor(v, 4, 32));
    v = fminf(v, __shfl_xor(v, 8, 32));
    acc[k] = v;
  }

  // VGPR k holds rows k (low half) and k+8 (high half).
  if (m == 0) {
    const int rbase = base + tile_i * 16 + (hi ? 8 : 0);
#pragma unroll
    for (int k = 0; k < 8; ++k) {
      float val = acc[k] + sqA[rbase + k];  // add |a_i|^2 after the min
      atomicMin(&out[rbase + k], f2key(val));
    }
  }
}

// Weighted means of the two min-vectors, then (a-b)^2.
__global__ __launch_bounds__(1024) void final_kernel(
    const unsigned* __restrict__ keys,
    const float* __restrict__ x_mask, const float* __restrict__ y_mask,
    float* __restrict__ outp) {
  __shared__ float sa[1024];
  __shared__ float sb[1024];
  const int t = threadIdx.x;
  float pa = 0.0f, pb = 0.0f;
  for (int i = t; i < BN; i += 1024) {
    pb += y_mask[i] * key2f(keys[i]);        // b = mean(y_mask * min_j P)  (dir 0)
    pa += x_mask[i] * key2f(keys[i + BN]);   // a = mean(x_mask * min_i P)  (dir 1)
  }
  sa[t] = pa; sb[t] = pb;
  __syncthreads();
  for (int s = 512; s > 0; s >>= 1) {
    if (t < s) { sa[t] += sa[t + s]; sb[t] += sb[t + s]; }
    __syncthreads();
  }
  if (t == 0) {
    float aM = sa[0] / (float)BN;
    float bM = sb[0] / (float)BN;
    outp[0] = (aM - bM) * (aM - bM);
  }
}

extern "C" void kernel_launch(void* const* d_in, const int* in_sizes, int n_in,
                              void* d_out, int out_size, void* d_ws, size_t ws_size,
                              hipStream_t stream) {
  const float* x  = (const float*)d_in[0];
  const float* y  = (const float*)d_in[1];
  const float* xm = (const float*)d_in[2];
  const float* ym = (const float*)d_in[3];

  float*    sqx  = (float*)d_ws;
  float*    sqy  = sqx + BN;
  unsigned* keys = (unsigned*)(sqy + BN);  // 2*BN uints; total ws = 512 KB

  prep_kernel<<<(BN + 255) / 256, 256, 0, stream>>>(x, y, sqx, sqy, keys);
  rowmin_kernel<<<dim3(TILES, CHUNKS, B * 2), 32, 0, stream>>>(x, y, sqx, sqy, keys);
  final_kernel<<<1, 1024, 0, stream>>>(keys, xm, ym, (float*)d_out);
}